// QuantumSpikingNeuronLayer_75703093559834
// MI455X (gfx1250) — compile-verified
//
#include <hip/hip_runtime.h>
#include <math.h>

#define NQ 14
#define NSTATE (1 << NQ)      // 16384 amplitudes
#define NQUAD  (NSTATE >> 2)  // 4096 quads per fused 2-wire pass
#define TPB 512
#define MOUT 8

__device__ __forceinline__ float2 cmul(float2 a, float2 b) {
    return make_float2(a.x * b.x - a.y * b.y, a.x * b.y + a.y * b.x);
}
__device__ __forceinline__ float2 cadd(float2 a, float2 b) {
    return make_float2(a.x + b.x, a.y + b.y);
}
__device__ __forceinline__ float2 cmac2(float2 u0, float2 a0, float2 u1, float2 a1) {
    return cadd(cmul(u0, a0), cmul(u1, a1));
}

// PennyLane Rot(phi, theta, omega) = RZ(omega) RY(theta) RZ(phi)
__device__ __forceinline__ void rot_mat(float phi, float th, float om, float2 U[2][2]) {
    float ct = cosf(0.5f * th), st = sinf(0.5f * th);
    float a  = 0.5f * (phi + om);
    float b2 = 0.5f * (phi - om);
    float2 ep = make_float2(cosf(a), -sinf(a));    // e^{-i(phi+omega)/2}
    float2 em = make_float2(cosf(b2), sinf(b2));   // e^{+i(phi-omega)/2}
    U[0][0] = make_float2( ep.x * ct,  ep.y * ct);
    U[0][1] = make_float2(-em.x * st, -em.y * st);
    U[1][0] = make_float2( em.x * st, -em.y * st); // conj(em)*st
    U[1][1] = make_float2( ep.x * ct, -ep.y * ct); // conj(ep)*ct
}

// Fused pass: gate U on flat bit bu and gate V on flat bit bv (commuting wires).
__device__ __forceinline__ void apply2q(float2* sA, const float2 U[2][2], int bu,
                                        const float2 V[2][2], int bv, int tid) {
    const int bl = bu < bv ? bu : bv;
    const int bh = bu < bv ? bv : bu;
    const unsigned ml = (1u << bl) - 1u;
    const unsigned mh = (1u << bh) - 1u;
    const unsigned iu = 1u << bu, iv = 1u << bv;
    for (int p = tid; p < NQUAD; p += TPB) {
        unsigned i = (unsigned)p;
        i = ((i >> bl) << (bl + 1)) | (i & ml);   // insert 0 at bl
        i = ((i >> bh) << (bh + 1)) | (i & mh);   // insert 0 at bh
        float2 a00 = sA[i];
        float2 a01 = sA[i | iv];
        float2 a10 = sA[i | iu];
        float2 a11 = sA[i | iu | iv];
        float2 b00 = cmac2(V[0][0], a00, V[0][1], a01);
        float2 b01 = cmac2(V[1][0], a00, V[1][1], a01);
        float2 b10 = cmac2(V[0][0], a10, V[0][1], a11);
        float2 b11 = cmac2(V[1][0], a10, V[1][1], a11);
        sA[i]           = cmac2(U[0][0], b00, U[0][1], b10);
        sA[i | iv]      = cmac2(U[0][0], b01, U[0][1], b11);
        sA[i | iu]      = cmac2(U[1][0], b00, U[1][1], b10);
        sA[i | iu | iv] = cmac2(U[1][0], b01, U[1][1], b11);
    }
}

// One workgroup per sample; statevector resident in LDS (~130 KB of 320 KB).
__global__ __launch_bounds__(TPB) void qsim_kernel(const float* __restrict__ x,
                                                   const float* __restrict__ qw,
                                                   float* __restrict__ zbuf) {
    __shared__ float2 sA[NSTATE];
    __shared__ float2 col0[NQ][2];     // column 0 of fused (Rot0 * RX) per wire
    __shared__ float2 prodLo[128];     // product over flat bits 0..6  (wires 13..7)
    __shared__ float2 prodHi[128];     // product over flat bits 7..13 (wires 6..0)
    const int b = blockIdx.x;
    const int tid = threadIdx.x;

    // Stage 1: per-wire fused gate first columns (|0> input selects column 0).
    if (tid < NQ) {
        int w = tid;
        float th = x[b * NQ + w];
        float cx = cosf(0.5f * th), sx = sinf(0.5f * th);
        float2 rx0 = make_float2(cx, 0.f);   // RX[0][0] = RX[1][1]... column 0
        float2 rx1 = make_float2(0.f, -sx);  // RX[1][0]
        float2 R[2][2];
        rot_mat(qw[w * 3 + 0], qw[w * 3 + 1], qw[w * 3 + 2], R);
        col0[w][0] = cmac2(R[0][0], rx0, R[0][1], rx1);
        col0[w][1] = cmac2(R[1][0], rx0, R[1][1], rx1);
    }
    __syncthreads();

    // Stage 2: 7-wire partial-product tables (product state factorizes).
    if (tid < 256) {
        int t = tid & 127;
        float2 prod = make_float2(1.f, 0.f);
        if (tid < 128) {
            #pragma unroll
            for (int f = 0; f < 7; ++f)                 // flat bit f -> wire 13-f
                prod = cmul(prod, col0[13 - f][(t >> f) & 1]);
            prodLo[t] = prod;
        } else {
            #pragma unroll
            for (int f = 0; f < 7; ++f)                 // flat bit 7+f -> wire 6-f
                prod = cmul(prod, col0[6 - f][(t >> f) & 1]);
            prodHi[t] = prod;
        }
    }
    __syncthreads();

    // Stage 3: ONE sweep builds the state after embedding + layer-0 Rot +
    // layer-0 CNOT ring (r=1). Ring is a GF(2)-linear basis permutation M0;
    // sA[j] = prodstate(M0^{-1}(j)), with
    //   i[0..11]  = (j ^ (j>>1)) & 0xFFF
    //   i[12]     = j13 ^ j12 ^ j0
    //   i[13]     = j13 ^ j0
    for (int j = tid; j < NSTATE; j += TPB) {
        unsigned uj = (unsigned)j;
        unsigned i = (uj ^ (uj >> 1)) & 0x0FFFu;
        i |= (((uj >> 12) ^ (uj >> 13) ^ uj) & 1u) << 12;
        i |= (((uj >> 13) ^ uj) & 1u) << 13;
        sA[j] = cmul(prodHi[i >> 7], prodLo[i & 127u]);
    }
    __syncthreads();

    // Stage 4: layer-1 Rot gates, two commuting wires fused per sweep.
    for (int w = 0; w < 7; ++w) {
        int w2 = w + 7;
        float2 U[2][2], V[2][2];
        rot_mat(qw[(NQ + w) * 3 + 0], qw[(NQ + w) * 3 + 1], qw[(NQ + w) * 3 + 2], U);
        rot_mat(qw[(NQ + w2) * 3 + 0], qw[(NQ + w2) * 3 + 1], qw[(NQ + w2) * 3 + 2], V);
        apply2q(sA, U, (NQ - 1) - w, V, (NQ - 1) - w2, tid);
        __syncthreads();
    }

    // Stage 5: <Z_w> with the layer-1 CNOT ring (r=2) folded in: ring maps
    // basis i -> M1(i); bit_w(M1(i)) is the parity of i under a fixed mask
    // (flat-bit space, wire w <-> bit 13-w):
    //   nb0=b2^b4^..^b12  nb1=b3^b5^..^b13  nb2=b0^b2  nb3=b1^b3
    //   nb4=b0^b2^b4      nb5=b1^b3^b5      nb6=b0..b6 even  nb7=b1..b7 odd
    const unsigned zmask[MOUT] = { 0x0AAAu, 0x0555u, 0x2800u, 0x1400u,
                                   0x2A00u, 0x1500u, 0x2A80u, 0x1540u };
    float z[MOUT];
    #pragma unroll
    for (int w = 0; w < MOUT; ++w) z[w] = 0.0f;
    for (int j = tid; j < NSTATE; j += TPB) {
        float2 a = sA[j];
        float p = a.x * a.x + a.y * a.y;
        #pragma unroll
        for (int w = 0; w < MOUT; ++w)
            z[w] += (__popc(((unsigned)j) & zmask[w]) & 1) ? -p : p;
    }
    __syncthreads();  // done reading state; reuse LDS as reduction scratch
    float* scratch = (float*)sA;
    #pragma unroll
    for (int w = 0; w < MOUT; ++w) scratch[w * TPB + tid] = z[w];
    __syncthreads();
    for (int s = TPB / 2; s > 0; s >>= 1) {
        if (tid < s) {
            #pragma unroll
            for (int w = 0; w < MOUT; ++w)
                scratch[w * TPB + tid] += scratch[w * TPB + tid + s];
        }
        __syncthreads();
    }
    if (tid < MOUT) zbuf[b * MOUT + tid] = scratch[tid * TPB];
}

// ---- head: sigmoid(z @ W^T + b) via V_WMMA_F32_16X16X4_F32 ----
typedef __attribute__((ext_vector_type(2))) float v2f;
typedef __attribute__((ext_vector_type(8))) float v8f;

__global__ __launch_bounds__(32) void head_kernel(const float* __restrict__ zbuf,
                                                  const float* __restrict__ W,
                                                  const float* __restrict__ bias,
                                                  float* __restrict__ out, int B) {
    const int lane = threadIdx.x;
    const int m    = lane & 15;   // A: M row / B & D: N col
    const int half = lane >> 4;
    const int row0 = blockIdx.x * 16;

    // Unconditional loads with clamped indices: out-of-range A rows only feed
    // D rows we never store; W cols n>=8 only feed D cols we never store.
    int rowA = row0 + m; if (rowA >= B) rowA = B - 1;
    int colB = m & (MOUT - 1);

    v8f acc = {};
    #pragma unroll
    for (int kOff = 0; kOff < 8; kOff += 4) {
        // A 16x4 f32: VGPR v, lane-half h -> K = kOff + 2h + v, M = lane&15
        int k0 = kOff + 2 * half;
        v2f a, bm;
        a.x  = zbuf[rowA * MOUT + k0 + 0];
        a.y  = zbuf[rowA * MOUT + k0 + 1];
        // B 4x16 f32 (K x N): B[k][n] = W[n][k]
        bm.x = W[colB * MOUT + k0 + 0];
        bm.y = W[colB * MOUT + k0 + 1];
        acc = __builtin_amdgcn_wmma_f32_16x16x4_f32(
            /*neg_a=*/false, a, /*neg_b=*/false, bm,
            /*c_mod=*/(short)0, acc, /*reuse_a=*/false, /*reuse_b=*/false);
    }
    // C/D 16x16: VGPR j -> M = j + 8*half, N = lane&15
    #pragma unroll
    for (int j = 0; j < 8; ++j) {
        int row = row0 + j + 8 * half;
        if (m < MOUT && row < B) {
            float v = acc[j] + bias[m];
            out[row * MOUT + m] = 1.0f / (1.0f + expf(-v));
        }
    }
}

extern "C" void kernel_launch(void* const* d_in, const int* in_sizes, int n_in,
                              void* d_out, int out_size, void* d_ws, size_t ws_size,
                              hipStream_t stream) {
    const float* x    = (const float*)d_in[0];  // (B, 14)
    const float* qw   = (const float*)d_in[1];  // (2, 14, 3)
    const float* W    = (const float*)d_in[2];  // (8, 8)
    const float* bias = (const float*)d_in[3];  // (8,)
    float* out  = (float*)d_out;                // (B, 8)
    float* zbuf = (float*)d_ws;                 // (B, 8) scratch

    const int B = in_sizes[0] / NQ;
    qsim_kernel<<<B, TPB, 0, stream>>>(x, qw, zbuf);
    const int nb = (B + 15) / 16;
    head_kernel<<<nb, 32, 0, stream>>>(zbuf, W, bias, out, B);
}